// ZigzagTransform_77970836292147
// MI455X (gfx1250) — compile-verified
//
#include <hip/hip_runtime.h>
#include <hip/hip_bf16.h>

// Zigzag channel permutation: y[b,p,c,:,:] = x[b,p,ZZ[c],:,:]
// x: (16, 3*64, 128, 128) fp32. Pure bandwidth-bound plane copy with a
// compile-time channel remap. Implemented via CDNA5 async global<->LDS DMA
// (GLOBAL_LOAD_ASYNC_TO_LDS_B128 / GLOBAL_STORE_ASYNC_FROM_LDS_B128,
// ASYNCcnt-tracked) so data never round-trips through VGPRs.

// Classic 8x8 zigzag scan order: PERM[k] = row-major index whose zigzag rank is k.
__constant__ int kZig[64] = {
     0,  1,  8, 16,  9,  2,  3, 10,
    17, 24, 32, 25, 18, 11,  4,  5,
    12, 19, 26, 33, 40, 48, 41, 34,
    27, 20, 13,  6,  7, 14, 21, 28,
    35, 42, 49, 56, 57, 50, 43, 36,
    29, 22, 15, 23, 30, 37, 44, 51,
    58, 59, 52, 45, 38, 31, 39, 46,
    53, 60, 61, 54, 47, 55, 62, 63
};

#define PLANE_ELEMS 16384u   // 128*128 floats per channel plane
#define CHUNK_ELEMS 4096u    // floats per block (16 KB)
#define SEG_BYTES   4096     // per-segment stride (both LDS and global)

// 4 async b128 loads -> s_wait_asynccnt 0 -> 4 async b128 stores.
// NOTE (CDNA5 ISA): the instruction OFFSET is added to BOTH the
// LDS address and the global address, and our LDS/global segment strides are
// both 4096 B, so one base address per direction suffices.
#define ASYNC_LD(OFF)                                                         \
    asm volatile("global_load_async_to_lds_b128 %0, %1, off offset:" #OFF     \
                 " th:TH_LOAD_NT"                                             \
                 :: "v"(lds_addr), "v"(src) : "memory")

#define ASYNC_ST(OFF)                                                         \
    asm volatile("global_store_async_from_lds_b128 %0, %1, off offset:" #OFF  \
                 " th:TH_STORE_NT"                                            \
                 :: "v"(dst), "v"(lds_addr) : "memory")

__global__ __launch_bounds__(256)
void ZigzagTransform_77970836292147_kernel(const float* __restrict__ in,
                                           float* __restrict__ out) {
    __shared__ alignas(16) char buf[16384];   // 4 segments x 4 KB

    const unsigned plane    = blockIdx.x;          // 0 .. nplanes-1 (b*192 + p*64 + c)
    const unsigned c        = plane & 63u;
    const unsigned srcPlane = (plane & ~63u) + (unsigned)kZig[c];
    const unsigned chunk    = blockIdx.y;          // 0..3
    const unsigned tid      = threadIdx.x;

    const float* src = in  + (size_t)srcPlane * PLANE_ELEMS
                           + (size_t)chunk * CHUNK_ELEMS + tid * 4u;
    float*       dst = out + (size_t)plane    * PLANE_ELEMS
                           + (size_t)chunk * CHUNK_ELEMS + tid * 4u;

    // Wave-relative LDS byte address of this lane's 16B slot: low 32 bits of
    // the generic (flat) pointer to the shared buffer.
    unsigned lds_addr = (unsigned)(unsigned long long)(void*)buf + tid * 16u;

    ASYNC_LD(0);
    ASYNC_LD(4096);
    ASYNC_LD(8192);
    ASYNC_LD(12288);

    asm volatile("s_wait_asynccnt 0x0" ::: "memory");

    ASYNC_ST(0);
    ASYNC_ST(4096);
    ASYNC_ST(8192);
    ASYNC_ST(12288);
    // S_ENDPGM performs an implicit wait-idle, draining the async stores.
}

extern "C" void kernel_launch(void* const* d_in, const int* in_sizes, int n_in,
                              void* d_out, int out_size, void* d_ws, size_t ws_size,
                              hipStream_t stream) {
    const float* x = (const float*)d_in[0];
    float*       y = (float*)d_out;

    const unsigned total   = (unsigned)in_sizes[0];          // 16*192*128*128
    const unsigned nplanes = total / PLANE_ELEMS;            // 3072
    const unsigned nchunks = PLANE_ELEMS / CHUNK_ELEMS;      // 4

    dim3 grid(nplanes, nchunks, 1);
    dim3 block(256, 1, 1);
    ZigzagTransform_77970836292147_kernel<<<grid, block, 0, stream>>>(x, y);
}